// CIN_44530220925272
// MI455X (gfx1250) — compile-verified
//
#include <hip/hip_runtime.h>
#include <hip/hip_bf16.h>

// ---------------------------------------------------------------------------
// DCN cross network on fp32 WMMA (v_wmma_f32_16x16x4_f32), wave32.
//   layer0: z0 = relu(f0^T vec(x⊗x))  M=B*D=65536, K=1521 (pad 40/row -> 1600), N=128
//   layer1: z1 = relu(f1^T vec(x⊗h))  M=65536,     K=2496,                      N=128
//   out[b] = sum_d sum_c concat(z0[64:],z1)[b,c,d]*dense_w[c] + dense_b
// AI ~5.6 kFLOP/B -> compute-bound; whole working set lives in the 192MB L2.
// One wave = one batch element (16 rows = d-slices); 8 waves/block.
// W slabs double-buffered in LDS in B-fragment order; next-next slab is
// global_prefetch_b8'd so staging loads hit near caches.
// ---------------------------------------------------------------------------

typedef __attribute__((ext_vector_type(2))) float v2f;
typedef __attribute__((ext_vector_type(8))) float v8f;

#define B_     4096
#define F0_    39
#define D_     16
#define LCH    128
#define K0_    1521          // true K of layer 0 (39*39)
#define K1_    2496          // layer 1 (39*64)
#define SLAB   64
#define NSLAB0 25            // padded K0 = 40*40 = 1600 = 25*64
#define NSLAB1 39            // 2496 = 39*64 exactly

#define XS_F   (128*41)      // x rows, stride 41 (odd -> conflict-free), [39],[40]=0
#define HS_F   (8*16*65)     // per-wave relu'd h rows (64 cols, stride 65)
#define WS_F   (SLAB*LCH)    // one swizzled W slab
#define SMEM_F (XS_F + HS_F + 2*WS_F)

// Padded layer-0 mapping kp = i*40 + j  ->  source row i*39 + min(j,38);
// padded cells are don't-care because the matching A elements are 0.
template <bool PAD40>
__device__ __forceinline__ int src_row(int kp, int kmax) {
  int ksrc;
  if (PAD40) {
    int i = kp / 40;
    int j = kp - i * 40;
    j     = j < 39 ? j : 38;
    ksrc  = i * 39 + j;
  } else {
    ksrc = kp;
  }
  return ksrc < kmax - 1 ? ksrc : kmax - 1;
}

// Prefetch one 64-row slab (one cacheline-touch per float4 chunk).
template <bool PAD40>
__device__ __forceinline__ void slab_prefetch(const float* __restrict__ W,
                                              int kb, int kmax, int tid) {
#pragma unroll
  for (int it = 0; it < 8; ++it) {
    int idx4  = tid + it * 256;
    int ksrc  = src_row<PAD40>(kb + (idx4 >> 5), kmax);
    int n0    = (idx4 & 31) << 2;
    __builtin_prefetch(W + (size_t)ksrc * LCH + n0, 0, 3);  // global_prefetch_b8
  }
}

// Stage one slab: global float4 loads (should hit near caches after prefetch)
// scattered into LDS in B-fragment order: fragment (kstep,ntile) is 32
// contiguous float2 by lane; value v of lane l is k=4*kstep+2*(l>>4)+v,
// n=ntile*16+(l&15).
template <bool PAD40>
__device__ __forceinline__ void slab_stage(const float* __restrict__ W, int kb,
                                           int kmax, float* __restrict__ ws,
                                           int tid) {
#pragma unroll
  for (int it = 0; it < 8; ++it) {
    int idx4  = tid + it * 256;
    int kroff = idx4 >> 5;
    int n0    = (idx4 & 31) << 2;
    int ksrc  = src_row<PAD40>(kb + kroff, kmax);
    const float4 v = *(const float4*)(W + (size_t)ksrc * LCH + n0);
    int kstep = kroff >> 2;
    int t     = kroff & 3;
    float vv[4] = {v.x, v.y, v.z, v.w};
#pragma unroll
    for (int c = 0; c < 4; ++c) {
      int n    = n0 + c;
      int lane = ((t >> 1) << 4) + (n & 15);
      ws[(((kstep << 3) + (n >> 4)) << 6) + lane * 2 + (t & 1)] = vv[c];
    }
  }
}

__global__ __launch_bounds__(256, 1) void cin_cross_kernel(
    const float* __restrict__ x, const float* __restrict__ f0,
    const float* __restrict__ f1, const float* __restrict__ dw,
    const float* __restrict__ db, float* __restrict__ out) {
  extern __shared__ float smem[];
  float* xs  = smem;                  // [128][41]
  float* hs  = smem + XS_F;           // [8][16][65]
  float* wsA = smem + XS_F + HS_F;    // slab buffer 0
  float* wsB = wsA + WS_F;            // slab buffer 1

  const int tid  = threadIdx.x;
  const int lane = tid & 31;
  const int wave = tid >> 5;
  const int half = lane >> 4;         // A layout: K offset = 2*half + v
  const int mrow = lane & 15;         // A layout: M row = lane&15
  const int b0   = blockIdx.x * 8;

  // ---- stage x rows: xs[w*16+d][i] = x[b0+w][i][d]; zero pad cols 39,40 ----
  for (int idx = tid; idx < 128 * F0_; idx += 256) {
    int m = idx / F0_, i = idx - m * F0_;
    int w = m >> 4, d = m & 15;
    xs[m * 41 + i] = x[(size_t)(b0 + w) * (F0_ * D_) + i * D_ + d];
  }
  { int m = tid >> 1; xs[m * 41 + 39 + (tid & 1)] = 0.0f; }

  const float* xrow = xs + (wave * 16 + mrow) * 41;
  float*       hrow = hs + wave * (16 * 65) + mrow * 65;

  v8f acc[8];
#pragma unroll
  for (int nt = 0; nt < 8; ++nt) acc[nt] = (v8f)0.0f;

  // =================== layer 0 : padded K = 1600, A = x[i]*x[j] =============
  slab_stage<true>(f0, 0, K0_, wsA, tid);
  slab_prefetch<true>(f0, SLAB, K0_, tid);
  __syncthreads();
  {
    int i0 = 0, j0 = 2 * half;
    for (int slab = 0; slab < NSLAB0; ++slab) {
      const v2f* wsv = (const v2f*)((slab & 1) ? wsB : wsA);
      if (slab + 2 < NSLAB0) slab_prefetch<true>(f0, (slab + 2) * SLAB, K0_, tid);
#pragma unroll 4
      for (int ks = 0; ks < 16; ++ks) {
        float xi  = xrow[i0];                  // zero in padded tail rows
        float xj0 = xrow[j0];
        float xj1 = xrow[j0 + 1];              // zero at padded col 39
        v2f A = {xi * xj0, xi * xj1};
        v2f Bv[8];
#pragma unroll
        for (int nt = 0; nt < 8; ++nt)         // batched: progressive dscnt waits
          Bv[nt] = wsv[((ks << 3) + nt) * 32 + lane];
#pragma unroll
        for (int nt = 0; nt < 8; ++nt)
          acc[nt] = __builtin_amdgcn_wmma_f32_16x16x4_f32(
              false, A, false, Bv[nt], (short)0, acc[nt], false, false);
        j0 += 4;
        int c = j0 >= 40;                      // branchless carry
        j0 -= 40 * c;
        i0 += c;
      }
      if (slab + 1 < NSLAB0)
        slab_stage<true>(f0, (slab + 1) * SLAB, K0_, (slab & 1) ? wsA : wsB, tid);
      __syncthreads();
    }
  }

  // relu; park h = z0[:, 0:64]; fold z0[:, 64:128] into the dense dot
  float partial = 0.0f;
#pragma unroll
  for (int nt = 0; nt < 4; ++nt) {
#pragma unroll
    for (int r = 0; r < 8; ++r) {              // C: row = r + 8*half, col = mrow
      float z = fmaxf(acc[nt][r], 0.0f);
      hs[wave * (16 * 65) + (r + 8 * half) * 65 + nt * 16 + mrow] = z;
    }
  }
#pragma unroll
  for (int nt = 4; nt < 8; ++nt) {
    float w = dw[nt * 16 + mrow - 64];
#pragma unroll
    for (int r = 0; r < 8; ++r) partial += fmaxf(acc[nt][r], 0.0f) * w;
  }

  // =================== layer 1 : K = 2496, A = x[i]*h[j] ====================
#pragma unroll
  for (int nt = 0; nt < 8; ++nt) acc[nt] = (v8f)0.0f;
  slab_stage<false>(f1, 0, K1_, wsA, tid);
  slab_prefetch<false>(f1, SLAB, K1_, tid);
  __syncthreads();
  {
    int i0 = 0, j0 = 2 * half;
    for (int slab = 0; slab < NSLAB1; ++slab) {
      const v2f* wsv = (const v2f*)((slab & 1) ? wsB : wsA);
      if (slab + 2 < NSLAB1) slab_prefetch<false>(f1, (slab + 2) * SLAB, K1_, tid);
#pragma unroll 4
      for (int ks = 0; ks < 16; ++ks) {
        float xi = xrow[i0];                   // i0 <= 38 always
        v2f A = {xi * hrow[j0], xi * hrow[j0 + 1]};
        v2f Bv[8];
#pragma unroll
        for (int nt = 0; nt < 8; ++nt)
          Bv[nt] = wsv[((ks << 3) + nt) * 32 + lane];
#pragma unroll
        for (int nt = 0; nt < 8; ++nt)
          acc[nt] = __builtin_amdgcn_wmma_f32_16x16x4_f32(
              false, A, false, Bv[nt], (short)0, acc[nt], false, false);
        j0 += 4;
        int c = j0 >= 64;
        j0 -= 64 * c;
        i0 += c;
      }
      if (slab + 1 < NSLAB1)
        slab_stage<false>(f1, (slab + 1) * SLAB, K1_, (slab & 1) ? wsA : wsB, tid);
      __syncthreads();
    }
  }

  // relu + dense dot for layer-1 channels (weight offset 64)
#pragma unroll
  for (int nt = 0; nt < 8; ++nt) {
    float w = dw[64 + nt * 16 + mrow];
#pragma unroll
    for (int r = 0; r < 8; ++r) partial += fmaxf(acc[nt][r], 0.0f) * w;
  }

  // wave32 reduction: this wave owns exactly one batch element
#pragma unroll
  for (int off = 16; off > 0; off >>= 1)
    partial += __shfl_xor(partial, off, 32);
  if (lane == 0) out[b0 + wave] = partial + db[0];
}

extern "C" void kernel_launch(void* const* d_in, const int* in_sizes, int n_in,
                              void* d_out, int out_size, void* d_ws,
                              size_t ws_size, hipStream_t stream) {
  const float* x  = (const float*)d_in[0];
  const float* f0 = (const float*)d_in[1];
  const float* f1 = (const float*)d_in[2];
  const float* dw = (const float*)d_in[3];
  const float* db = (const float*)d_in[4];
  float* out = (float*)d_out;
  (void)in_sizes; (void)n_in; (void)out_size; (void)d_ws; (void)ws_size;

  dim3 grid(B_ / 8);          // 512 blocks; 8 waves; 1 wave per batch element
  dim3 block(256);
  size_t smem = SMEM_F * sizeof(float);   // ~117 KB of 320 KB WGP LDS
  cin_cross_kernel<<<grid, block, smem, stream>>>(x, f0, f1, dw, db, out);
}